// DecouplePreAggGraphConv_35192962023437
// MI455X (gfx1250) — compile-verified
//
#include <hip/hip_runtime.h>

#define BB 16384
#define JJ 17
#define CC 128            // FIN == FOUT
#define BT 32             // b-rows per workgroup
#define XS_STRIDE 132     // dwords per staged x row (128 + 4 pad -> conflict-free A loads)
#define H1_STRIDE 2180    // dwords per h1 row (17*128 + 4 pad)

typedef float v2f __attribute__((ext_vector_type(2)));
typedef float v4f __attribute__((ext_vector_type(4)));
typedef float v8f __attribute__((ext_vector_type(8)));
typedef unsigned int u32x4 __attribute__((ext_vector_type(4)));
typedef int i32x4 __attribute__((ext_vector_type(4)));
typedef int i32x8 __attribute__((ext_vector_type(8)));

// ---------------------------------------------------------------------------
// Repack W[2][17][128][128] (row-major k x n) into WMMA B-fragment order for
// V_WMMA_F32_16X16X4_F32, and fold diag(adj) into the g=0 (self) weights.
// B frag (4x16, K x N): VGPR0: lanes 0-15 = (K=4ks+0, N=l), lanes 16-31 =
// (K=4ks+2, N=l-16); VGPR1: K+1 / K+3.  Packed so each lane loads its 2
// floats with one coalesced b64: offset = frag_base + lane*2 + vgpr.
// ---------------------------------------------------------------------------
__global__ void pack_w_kernel(const float* __restrict__ W,
                              const float* __restrict__ adj,
                              float* __restrict__ WP) {
  int idx = blockIdx.x * blockDim.x + threadIdx.x;
  const int total = 2 * JJ * CC * CC;
  if (idx >= total) return;
  int n = idx & 127;
  int k = (idx >> 7) & 127;
  int j = (idx >> 14) % JJ;
  int g = idx / (JJ * CC * CC);
  float w = W[idx];
  if (g == 0) w *= adj[j * JJ + j];        // fold diagonal scaling into W0
  int nt = n >> 4, nn = n & 15, ks = k >> 2, r = k & 3;
  int lane = (r < 2) ? nn : (16 + nn);
  int vg = r & 1;
  long o = ((((long)(g * JJ + j) * 8 + nt) * 32 + ks) * 64) + lane * 2 + vg;
  WP[o] = w;
}

// ---------------------------------------------------------------------------
// Tensor Data Mover: DMA one x tile x[b0..b0+31, j, 0..127] (fp32) into LDS,
// inserting 4 pad dwords after every 128 dwords so the LDS image has the
// conflict-free XS_STRIDE=132 row pitch.  Tracked by TENSORcnt.
//   D# group0: count=1 | lds_addr | global_addr[56:0] | type=2
//   D# group1: data_size=4B, pad_enable, pad_interval=128dw, pad_amount=4dw,
//              tensor 128x32, tile 128x32, dim0 stride = 2176 elements.
// ---------------------------------------------------------------------------
__device__ __forceinline__ void tdm_stage_x(const float* __restrict__ gsrc,
                                            unsigned lds_byte_off) {
  unsigned long long ga = (unsigned long long)(const void*)gsrc;
  u32x4 g0;
  g0[0] = 1u;                                            // count=1 (valid D#)
  g0[1] = lds_byte_off;                                  // lds_addr (bytes)
  g0[2] = (unsigned)(ga & 0xffffffffu);                  // global_addr[31:0]
  g0[3] = (unsigned)((ga >> 32) & 0x01ffffffu)           // global_addr[56:32]
        | (2u << 30);                                    // type = 2 ("image")
  i32x8 g1;
  g1[0] = (2 << 16)          // data_size: 4 bytes
        | (1 << 20)          // pad_enable
        | (6 << 22)          // pad_interval: 128 dwords
        | (3 << 25);         // pad_amount: 4 dwords
  g1[1] = (CC << 16);        // tensor_dim0[15:0] = 128
  g1[2] = (BT << 16);        // tensor_dim0[31:16]=0 | tensor_dim1[15:0] = 32
  g1[3] = (CC << 16);        // tensor_dim1[31:16]=0 | tile_dim0 = 128
  g1[4] = BT;                // tile_dim1 = 32, tile_dim2 = 0
  g1[5] = JJ * CC;           // tensor_dim0_stride[31:0] = 2176 elements
  g1[6] = 0;                 // stride0[47:32]=0 | tensor_dim1_stride[15:0]=0
  g1[7] = 0;
  i32x4 z4 = {0, 0, 0, 0};
#if defined(__clang_major__) && (__clang_major__ >= 23)
  i32x8 z8 = {0, 0, 0, 0, 0, 0, 0, 0};
  __builtin_amdgcn_tensor_load_to_lds(g0, g1, z4, z4, z8, 0);
#else
  __builtin_amdgcn_tensor_load_to_lds(g0, g1, z4, z4, 0);
#endif
}

// ---------------------------------------------------------------------------
// Fused kernel: one WG owns 32 b-rows.
//   phase 0 (g=1): per joint j, TDM-stage x[:,j,:], h1 = x @ W1[j] via f32
//                  WMMA, scatter into LDS h1 buffer.
//   phase 1 (g=0): TDM-stage x again, h0d = x @ (diag*W0[j]) via WMMA, add
//                  off[j,:] @ h1 cross term from LDS plus bias, store out.
// x staging is double-buffered: the TDM for joint j+1 runs under the WMMA
// work of joint j; everyone syncs on s_wait_tensorcnt + barrier.
// ---------------------------------------------------------------------------
__global__ void __launch_bounds__(256, 1)
gconv_kernel(const float* __restrict__ x,
             const float* __restrict__ bias,
             const float* __restrict__ adj,
             const float* __restrict__ WP,
             float* __restrict__ out) {
  extern __shared__ float lds[];
  float* xs0 = lds;                            // BT * XS_STRIDE (buffer 0)
  float* xs1 = lds + BT * XS_STRIDE;           // BT * XS_STRIDE (buffer 1)
  float* h1  = lds + 2 * BT * XS_STRIDE;       // BT * H1_STRIDE
  const unsigned XS_BYTES = BT * XS_STRIDE * 4u;

  const int tid  = threadIdx.x;
  const int wave = tid >> 5;
  const int lane = tid & 31;
  const int lhi  = lane >> 4;                  // 0: lanes 0-15, 1: lanes 16-31
  const int llo  = lane & 15;
  const long b0  = (long)blockIdx.x * BT;
  const float* xg = x + b0 * (JJ * CC);        // this WG's x tile base

  for (int phase = 0; phase < 2; ++phase) {
    const int g = 1 - phase;                   // neighbor weights first

    // prologue: DMA joint 0 into buffer 0
    if (wave == 0) tdm_stage_x(xg, 0u);
    __builtin_amdgcn_s_wait_tensorcnt((short)0);
    __syncthreads();

    for (int j = 0; j < JJ; ++j) {
      // kick off DMA for the next joint into the other buffer
      if (j + 1 < JJ && wave == 0)
        tdm_stage_x(xg + (j + 1) * CC, ((unsigned)(j + 1) & 1u) * XS_BYTES);

      const float* xs = (j & 1) ? xs1 : xs0;

      // ---- per-wave GEMM: [32 x 128] @ [128 x 16] n-tile, K steps of 4 ----
      const int nt = wave;
      v8f c0 = {};                             // rows 0-15
      v8f c1 = {};                             // rows 16-31
      const float* wp    = WP + (((long)(g * JJ + j) * 8 + nt) * 32) * 64;
      const float* arow0 = xs + llo * XS_STRIDE + 2 * lhi;   // A frag base
      const float* arow1 = arow0 + 16 * XS_STRIDE;
#pragma unroll 8
      for (int ks = 0; ks < 32; ++ks) {
        v2f b  = *(const v2f*)(wp + ks * 64 + lane * 2);     // packed W, b64
        v2f a0 = *(const v2f*)(arow0 + ks * 4);              // ds b64
        v2f a1 = *(const v2f*)(arow1 + ks * 4);
        c0 = __builtin_amdgcn_wmma_f32_16x16x4_f32(false, a0, false, b,
                                                   (short)0, c0, false, false);
        c1 = __builtin_amdgcn_wmma_f32_16x16x4_f32(false, a1, false, b,
                                                   (short)0, c1, false, false);
      }

      if (g == 1) {
        // ---- scatter h1 tile into LDS ([row][j][m], padded stride) ----
        const int m = nt * 16 + llo;
#pragma unroll
        for (int v = 0; v < 8; ++v) {
          int r0 = v + 8 * lhi;
          h1[r0 * H1_STRIDE + j * CC + m]        = c0[v];
          h1[(16 + r0) * H1_STRIDE + j * CC + m] = c1[v];
        }
      } else {
        // ---- epilogue: out = h0*diag (folded) + off[j,:] @ h1 + bias ----
        float offr[JJ];
#pragma unroll
        for (int k = 0; k < JJ; ++k)
          offr[k] = (k == j) ? 0.0f : adj[j * JJ + k];
        const int   m  = nt * 16 + llo;
        const float bv = bias[m];
#pragma unroll
        for (int v = 0; v < 8; ++v) {
#pragma unroll
          for (int mt = 0; mt < 2; ++mt) {
            int r = mt * 16 + v + 8 * lhi;
            float s = (mt ? c1[v] : c0[v]) + bv;
            const float* hp = h1 + r * H1_STRIDE + m;
#pragma unroll
            for (int k = 0; k < JJ; ++k)
              s = fmaf(offr[k], hp[k * CC], s);
            out[(b0 + r) * (JJ * CC) + (long)j * CC + m] = s;
          }
        }
      }

      // next-joint DMA must be complete and everyone done with buffers
      __builtin_amdgcn_s_wait_tensorcnt((short)0);
      __syncthreads();
    }
  }
}

extern "C" void kernel_launch(void* const* d_in, const int* in_sizes, int n_in,
                              void* d_out, int out_size, void* d_ws, size_t ws_size,
                              hipStream_t stream) {
  const float* x    = (const float*)d_in[0];
  const float* W    = (const float*)d_in[1];
  const float* bias = (const float*)d_in[2];
  const float* adj  = (const float*)d_in[3];
  float* out = (float*)d_out;
  float* WP  = (float*)d_ws;             // 2*17*128*128 floats = 2.23 MB

  const int totalW = 2 * JJ * CC * CC;
  pack_w_kernel<<<(totalW + 255) / 256, 256, 0, stream>>>(W, adj, WP);

  const size_t shmem =
      (size_t)(2 * BT * XS_STRIDE + BT * H1_STRIDE) * sizeof(float);
  gconv_kernel<<<dim3(BB / BT), 256, shmem, stream>>>(x, bias, adj, WP, out);
}